// G2Dist_PNAConv_20469814133065
// MI455X (gfx1250) — compile-verified
//
#include <hip/hip_runtime.h>

#define N_    10000
#define E_    160000
#define NG_   16
#define T_    5
#define F_    80
#define FOUT_ 16
#define C13_  (13 * F_)   // 1040
#define TF_   (T_ * F_)   // 400
#define AVG_DEG_LOG_ 2.8332133440562162f   // log(17)
#define BN_EPS_  1e-5f
#define STD_EPS_ 1e-5f

typedef float v2f __attribute__((ext_vector_type(2)));
typedef float v8f __attribute__((ext_vector_type(8)));

// ---------------------------------------------------------------------------
// Embedding gather: h[n, c] = emb[x[n, c/2]*2 + c%2]
// ---------------------------------------------------------------------------
__global__ void embed_kernel(const int* __restrict__ x, const float* __restrict__ emb,
                             float* __restrict__ h) {
  int i = blockIdx.x * blockDim.x + threadIdx.x;
  if (i >= N_ * F_) return;
  int n = i / F_, c = i % F_;
  int k = c >> 1, j = c & 1;
  h[i] = emb[x[n * 40 + k] * 2 + j];
}

// ---------------------------------------------------------------------------
// CSR build
// ---------------------------------------------------------------------------
__global__ void zero_u32_kernel(unsigned int* __restrict__ p, int n) {
  int i = blockIdx.x * blockDim.x + threadIdx.x;
  if (i < n) p[i] = 0u;
}

__global__ void copy_i32_kernel(const int* __restrict__ s, int* __restrict__ d, int n) {
  int i = blockIdx.x * blockDim.x + threadIdx.x;
  if (i < n) d[i] = s[i];
}

__global__ void count_deg_kernel(const int* __restrict__ dst, int* __restrict__ deg) {
  int e = blockIdx.x * blockDim.x + threadIdx.x;
  if (e < E_) atomicAdd(&deg[dst[e]], 1);
}

// Exclusive scan of deg[0..N) into row_start[0..N]; single block of 1024.
__global__ __launch_bounds__(1024) void scan_kernel(const int* __restrict__ deg,
                                                    int* __restrict__ row_start) {
  __shared__ int buf[1024];
  __shared__ int carry;
  if (threadIdx.x == 0) carry = 0;
  __syncthreads();
  for (int base = 0; base < N_; base += 1024) {
    int i = base + threadIdx.x;
    int v = (i < N_) ? deg[i] : 0;
    buf[threadIdx.x] = v;
    __syncthreads();
    for (int off = 1; off < 1024; off <<= 1) {
      int t = (threadIdx.x >= off) ? buf[threadIdx.x - off] : 0;
      __syncthreads();
      buf[threadIdx.x] += t;
      __syncthreads();
    }
    if (i < N_) row_start[i] = carry + buf[threadIdx.x] - v;  // exclusive
    int total = buf[1023];
    __syncthreads();
    if (threadIdx.x == 0) carry += total;
    __syncthreads();
  }
  if (threadIdx.x == 0) row_start[N_] = carry;
}

__global__ void fill_csr_kernel(const int* __restrict__ src, const int* __restrict__ dst,
                                int* __restrict__ cursor, int* __restrict__ csr_src) {
  int e = blockIdx.x * blockDim.x + threadIdx.x;
  if (e >= E_) return;
  int p = atomicAdd(&cursor[dst[e]], 1);
  csr_src[p] = src[e];
}

// Insertion-sort each node's neighbor list by value -> deterministic fp order.
__global__ void sort_rows_kernel(const int* __restrict__ row_start, int* __restrict__ csr_src) {
  int i = blockIdx.x * blockDim.x + threadIdx.x;
  if (i >= N_) return;
  int a = row_start[i], b = row_start[i + 1];
  for (int j = a + 1; j < b; ++j) {
    int key = csr_src[j];
    int k = j - 1;
    while (k >= a && csr_src[k] > key) { csr_src[k + 1] = csr_src[k]; --k; }
    csr_src[k + 1] = key;
  }
}

// ---------------------------------------------------------------------------
// Pre-projection GEMM (fp32 WMMA 16x16x4).
// Computes hi[n,t,g] = sum_f h[n,f]*pre_w[t,g,f]       (output cols 0..79)
//          hj[n,t,g] = sum_f h[n,f]*pre_w[t,g,80+f]    (output cols 80..159)
// One wave per 16x16 tile; grid = (625 M-tiles, 5 towers * 10 N-tiles).
// ---------------------------------------------------------------------------
__global__ __launch_bounds__(32) void pre_gemm_kernel(const float* __restrict__ h,
                                                      const float* __restrict__ pre_w,
                                                      float* __restrict__ hi,
                                                      float* __restrict__ hj) {
  const int m0   = blockIdx.x * 16;
  const int t    = blockIdx.y / 10;
  const int col0 = (blockIdx.y % 10) * 16;
  const int lane = threadIdx.x;
  const int half = lane >> 4;   // 0: K pair {0,1}; 1: K pair {2,3}
  const int l    = lane & 15;
  const int gc   = col0 + l;    // output column 0..159
  const float* wt = pre_w + t * (F_ * 2 * F_);
  // B[k][gc]: hi part -> wt[gc*160 + k]; hj part -> wt[(gc-80)*160 + 80 + k]
  const float* wbase = (gc < F_) ? (wt + gc * 160) : (wt + (gc - F_) * 160 + F_);
  const float* arow  = h + (m0 + l) * F_;
  v8f c = {};
  for (int k0 = 0; k0 < F_; k0 += 4) {
    int ka = k0 + 2 * half;
    v2f a, b;
    a.x = arow[ka];     a.y = arow[ka + 1];
    b.x = wbase[ka];    b.y = wbase[ka + 1];
    c = __builtin_amdgcn_wmma_f32_16x16x4_f32(false, a, false, b, (short)0, c, false, false);
  }
  for (int v = 0; v < 8; ++v) {
    int row = m0 + v + half * 8;
    if (gc < F_) hi[row * TF_ + t * F_ + gc]        = c[v];
    else         hj[row * TF_ + t * F_ + (gc - F_)] = c[v];
  }
}

// ---------------------------------------------------------------------------
// Fused aggregation + degree scalers + per-tower post-MLP matvec.
// One block (256 threads) per node. 13F*T vector staged in LDS (20.8 KB).
// ---------------------------------------------------------------------------
__global__ __launch_bounds__(256) void agg_post_kernel(
    const float* __restrict__ h, const float* __restrict__ hi, const float* __restrict__ hj,
    const float* __restrict__ pre_b, const float* __restrict__ post_w,
    const float* __restrict__ post_b, const int* __restrict__ row_start,
    const int* __restrict__ csr_src, float* __restrict__ postout) {
  __shared__ float l13[T_ * C13_];  // 5200 floats
  const int n   = blockIdx.x;
  const int tid = threadIdx.x;
  const int r0 = row_start[n], r1 = row_start[n + 1];
  const int d  = r1 - r0;
  const float degc = (float)((d > 1) ? d : 1);
  const float amp  = logf(degc + 1.0f) / AVG_DEG_LOG_;
  const float ramp = 1.0f / amp;

  // ht section (replicated per tower)
  for (int c = tid; c < TF_; c += 256) {
    int t = c / F_, g = c % F_;
    l13[t * C13_ + g] = h[n * F_ + g];
  }
  // per-channel edge aggregation
  for (int c = tid; c < TF_; c += 256) {
    int t = c / F_, g = c % F_;
    float base = hi[n * TF_ + c] + pre_b[c];
    float s = 0.f, s2 = 0.f;
    float mn = 3.402823466e38f, mx = -3.402823466e38f;
    for (int e = r0; e < r1; ++e) {
      int sidx = csr_src[e];
      float v = base + hj[sidx * TF_ + c];
      s += v; s2 += v * v;
      mn = fminf(mn, v); mx = fmaxf(mx, v);
    }
    float mean  = s  / degc;
    float mean2 = s2 / degc;
    float stdv  = sqrtf(fmaxf(mean2 - mean * mean, 0.f) + STD_EPS_);
    if (d == 0) { mean = 0.f; mn = 0.f; mx = 0.f; stdv = sqrtf(STD_EPS_); }
    float* L = l13 + t * C13_;
    L[ 1 * F_ + g] = mean;        L[ 2 * F_ + g] = mn;
    L[ 3 * F_ + g] = mx;          L[ 4 * F_ + g] = stdv;
    L[ 5 * F_ + g] = mean * amp;  L[ 6 * F_ + g] = mn * amp;
    L[ 7 * F_ + g] = mx * amp;    L[ 8 * F_ + g] = stdv * amp;
    L[ 9 * F_ + g] = mean * ramp; L[10 * F_ + g] = mn * ramp;
    L[11 * F_ + g] = mx * ramp;   L[12 * F_ + g] = stdv * ramp;
  }
  __syncthreads();

  // post-MLP: 80 outputs (5 towers x 16); one wave per output, 10 per wave.
  const int wave = tid >> 5, lane = tid & 31;
  for (int oi = 0; oi < 10; ++oi) {
    int o = wave + 8 * oi;              // 0..79
    int t = o >> 4, oo = o & 15;
    const float* W = post_w + (t * FOUT_ + oo) * C13_;
    const float* L = l13 + t * C13_;
    float acc = 0.f;
    for (int c2 = lane; c2 < C13_; c2 += 32) acc += L[c2] * W[c2];
    for (int off = 16; off; off >>= 1) acc += __shfl_down(acc, off, 32);
    if (lane == 0) postout[n * (T_ * FOUT_) + o] = acc + post_b[o];
  }
}

// ---------------------------------------------------------------------------
// lin GEMM (fp32 WMMA): y[n,o] = sum_c p[n,c]*w[o,c] + b[o]   (80x80)
// ---------------------------------------------------------------------------
__global__ __launch_bounds__(32) void lin_gemm_kernel(const float* __restrict__ p,
                                                      const float* __restrict__ w,
                                                      const float* __restrict__ bias,
                                                      float* __restrict__ y) {
  const int m0   = blockIdx.x * 16;
  const int col0 = blockIdx.y * 16;
  const int lane = threadIdx.x;
  const int half = lane >> 4;
  const int l    = lane & 15;
  const int o    = col0 + l;
  const float* arow = p + (m0 + l) * F_;
  const float* wrow = w + o * F_;
  v8f c = {};
  for (int k0 = 0; k0 < F_; k0 += 4) {
    int ka = k0 + 2 * half;
    v2f a, b;
    a.x = arow[ka];  a.y = arow[ka + 1];
    b.x = wrow[ka];  b.y = wrow[ka + 1];
    c = __builtin_amdgcn_wmma_f32_16x16x4_f32(false, a, false, b, (short)0, c, false, false);
  }
  for (int v = 0; v < 8; ++v) {
    int row = m0 + v + half * 8;
    y[row * F_ + o] = c[v] + bias[o];
  }
}

// ---------------------------------------------------------------------------
// BatchNorm (training-mode, biased var) + ReLU
// ---------------------------------------------------------------------------
__global__ __launch_bounds__(256) void bn_stats_kernel(const float* __restrict__ y,
                                                       const float* __restrict__ g,
                                                       float* __restrict__ mu,
                                                       float* __restrict__ rs) {
  __shared__ float ss[256], ss2[256];
  int c = blockIdx.x, tid = threadIdx.x;
  float s = 0.f, s2 = 0.f;
  for (int n = tid; n < N_; n += 256) {
    float v = y[n * F_ + c];
    s += v; s2 += v * v;
  }
  ss[tid] = s; ss2[tid] = s2;
  __syncthreads();
  for (int off = 128; off; off >>= 1) {
    if (tid < off) { ss[tid] += ss[tid + off]; ss2[tid] += ss2[tid + off]; }
    __syncthreads();
  }
  if (tid == 0) {
    float m   = ss[0] / (float)N_;
    float var = ss2[0] / (float)N_ - m * m;
    mu[c] = m;
    rs[c] = g[c] * rsqrtf(fmaxf(var, 0.f) + BN_EPS_);
  }
}

__global__ void bn_apply_relu_kernel(const float* __restrict__ y, const float* __restrict__ mu,
                                     const float* __restrict__ rs, const float* __restrict__ bb,
                                     float* __restrict__ hout) {
  int i = blockIdx.x * blockDim.x + threadIdx.x;
  if (i >= N_ * F_) return;
  int c = i % F_;
  hout[i] = fmaxf(0.f, (y[i] - mu[c]) * rs[c] + bb[c]);
}

// ---------------------------------------------------------------------------
// Global max pool: ReLU outputs are >= 0 so uint atomicMax on float bits is an
// exact, order-free max; 0-init matches "empty graph -> 0".
// ---------------------------------------------------------------------------
__global__ void pool_max_kernel(const float* __restrict__ h, const int* __restrict__ batch,
                                unsigned int* __restrict__ pooled) {
  int i = blockIdx.x * blockDim.x + threadIdx.x;
  if (i >= N_ * F_) return;
  int n = i / F_, c = i % F_;
  atomicMax(&pooled[batch[n] * F_ + c], __float_as_uint(h[i]));
}

__global__ void fc_kernel(const unsigned int* __restrict__ pooled, const float* __restrict__ fw,
                          const float* __restrict__ fb, float* __restrict__ out) {
  int i = blockIdx.x * blockDim.x + threadIdx.x;
  if (i >= NG_ * 20) return;
  int gi = i / 20, o = i % 20;
  float acc = fb[o];
  for (int c = 0; c < F_; ++c)
    acc += __uint_as_float(pooled[gi * F_ + c]) * fw[o * F_ + c];
  out[i] = acc;
}

// ---------------------------------------------------------------------------
extern "C" void kernel_launch(void* const* d_in, const int* in_sizes, int n_in,
                              void* d_out, int out_size, void* d_ws, size_t ws_size,
                              hipStream_t stream) {
  const int*   x        = (const int*)d_in[0];
  const int*   edge     = (const int*)d_in[1];   // [2,E]: src = edge, dst = edge+E
  const int*   batch    = (const int*)d_in[2];
  const float* emb      = (const float*)d_in[3];
  const float* fc_w     = (const float*)d_in[20];
  const float* fc_b     = (const float*)d_in[21];

  const int* srcp = edge;
  const int* dstp = edge + E_;

  // workspace carve-out (256B aligned)
  char* w = (char*)d_ws;
  auto alloc = [&](size_t bytes) -> void* {
    void* p = (void*)w;
    w += (bytes + 255) & ~(size_t)255;
    return p;
  };
  float* hA        = (float*)alloc(sizeof(float) * N_ * F_);
  float* hB        = (float*)alloc(sizeof(float) * N_ * F_);
  float* hi        = (float*)alloc(sizeof(float) * N_ * TF_);
  float* hj        = (float*)alloc(sizeof(float) * N_ * TF_);
  float* postout   = (float*)alloc(sizeof(float) * N_ * F_);
  float* ybuf      = (float*)alloc(sizeof(float) * N_ * F_);
  int*   deg       = (int*)alloc(sizeof(int) * N_);
  int*   row_start = (int*)alloc(sizeof(int) * (N_ + 1));
  int*   cursor    = (int*)alloc(sizeof(int) * N_);
  int*   csr_src   = (int*)alloc(sizeof(int) * E_);
  float* mu        = (float*)alloc(sizeof(float) * F_);
  float* rs        = (float*)alloc(sizeof(float) * F_);
  unsigned int* pooled = (unsigned int*)alloc(sizeof(unsigned int) * NG_ * F_);

  const int TPB = 256;
  const int gNF = (N_ * F_ + TPB - 1) / TPB;
  const int gE  = (E_ + TPB - 1) / TPB;
  const int gN  = (N_ + TPB - 1) / TPB;

  // init + embedding
  zero_u32_kernel<<<gN, TPB, 0, stream>>>((unsigned int*)deg, N_);
  zero_u32_kernel<<<(NG_ * F_ + TPB - 1) / TPB, TPB, 0, stream>>>(pooled, NG_ * F_);
  embed_kernel<<<gNF, TPB, 0, stream>>>(x, emb, hA);

  // CSR build (deterministic after per-row sort)
  count_deg_kernel<<<gE, TPB, 0, stream>>>(dstp, deg);
  scan_kernel<<<1, 1024, 0, stream>>>(deg, row_start);
  copy_i32_kernel<<<gN, TPB, 0, stream>>>(row_start, cursor, N_);
  fill_csr_kernel<<<gE, TPB, 0, stream>>>(srcp, dstp, cursor, csr_src);
  sort_rows_kernel<<<gN, TPB, 0, stream>>>(row_start, csr_src);

  float* hin  = hA;
  float* hout = hB;
  for (int l = 0; l < 2; ++l) {
    const float* pre_w  = (const float*)d_in[4 + 8 * l + 0];
    const float* pre_b  = (const float*)d_in[4 + 8 * l + 1];
    const float* post_w = (const float*)d_in[4 + 8 * l + 2];
    const float* post_b = (const float*)d_in[4 + 8 * l + 3];
    const float* lin_w  = (const float*)d_in[4 + 8 * l + 4];
    const float* lin_b  = (const float*)d_in[4 + 8 * l + 5];
    const float* bn_g   = (const float*)d_in[4 + 8 * l + 6];
    const float* bn_b   = (const float*)d_in[4 + 8 * l + 7];

    pre_gemm_kernel<<<dim3(N_ / 16, T_ * 10), 32, 0, stream>>>(hin, pre_w, hi, hj);
    agg_post_kernel<<<N_, 256, 0, stream>>>(hin, hi, hj, pre_b, post_w, post_b,
                                            row_start, csr_src, postout);
    lin_gemm_kernel<<<dim3(N_ / 16, F_ / 16), 32, 0, stream>>>(postout, lin_w, lin_b, ybuf);
    bn_stats_kernel<<<F_, 256, 0, stream>>>(ybuf, bn_g, mu, rs);
    bn_apply_relu_kernel<<<gNF, TPB, 0, stream>>>(ybuf, mu, rs, bn_b, hout);

    float* tmp = hin; hin = hout; hout = tmp;
  }

  // hin now holds final node features
  pool_max_kernel<<<gNF, TPB, 0, stream>>>(hin, batch, pooled);
  fc_kernel<<<(NG_ * 20 + TPB - 1) / TPB, TPB, 0, stream>>>(pooled, fc_w, fc_b, (float*)d_out);
}